// SimpleGNN_4002909520620
// MI455X (gfx1250) — compile-verified
//
#include <hip/hip_runtime.h>
#include <hip/hip_bf16.h>

#define N_NODES 50000
#define E_EDGES 800000
#define EMBED   64
#define FEAT    256
#define HIDDEN  128

typedef __attribute__((ext_vector_type(2))) float v2f;
typedef __attribute__((ext_vector_type(8))) float v8f;

// D = A(16x4) * B(4x16) + C, exact fp32 per-wave matrix op.
__device__ __forceinline__ v8f wmma_f32_k4(v2f a, v2f b, v8f c) {
    return __builtin_amdgcn_wmma_f32_16x16x4_f32(
        /*neg_a=*/false, a, /*neg_b=*/false, b,
        /*c_mod=*/(short)0, c, /*reuse_a=*/false, /*reuse_b=*/false);
}

// ---------------------------------------------------------------------------
// Kernel 1: feature encoder.  One wave per 16-node tile.
//   num_embed = relu(x_num @ Wnum + bnum)           (16x64)
//   feat      = [num_embed | e0 | e1 | e2]          (16x256) in LDS
//   h         = relu(feat @ Win + bin)              (16x128) -> global
// ---------------------------------------------------------------------------
__global__ __launch_bounds__(32) void gnn_encode_kernel(
    const float* __restrict__ x_num, const int* __restrict__ x_cat,
    const float* __restrict__ w_num, const float* __restrict__ b_num,
    const float* __restrict__ emb0, const float* __restrict__ emb1,
    const float* __restrict__ emb2,
    const float* __restrict__ w_in, const float* __restrict__ b_in,
    float* __restrict__ h)
{
    __shared__ float feat[16][FEAT];
    const int lane  = threadIdx.x;
    const int base  = blockIdx.x * 16;
    const int row16 = lane & 15;          // A row / B column / C column (in tile)
    const int khalf = (lane >> 4) * 2;    // lanes 16-31 carry K+2,K+3
    const int crow0 = (lane >> 4) * 8;    // C: lanes 16-31 carry rows 8-15

    // ---- GEMM1: x_num(16x64) @ w_num(64x64) -> feat[:,0:64]
    for (int nt = 0; nt < 4; ++nt) {
        v8f acc = {};
        const int ncol = nt * 16 + row16;
        for (int k = 0; k < EMBED; k += 4) {
            v2f a, b;
            a.x = x_num[(size_t)(base + row16) * 64 + k + khalf];
            a.y = x_num[(size_t)(base + row16) * 64 + k + khalf + 1];
            b.x = w_num[(k + khalf) * 64 + ncol];
            b.y = w_num[(k + khalf + 1) * 64 + ncol];
            acc = wmma_f32_k4(a, b, acc);
        }
        const float bias = b_num[ncol];
#pragma unroll
        for (int v = 0; v < 8; ++v) {
            float val = acc[v] + bias;
            feat[crow0 + v][ncol] = val > 0.f ? val : 0.f;
        }
    }

    // ---- gather 3 embedding tables -> feat[:,64:256]  (16*3*64 = 3072 floats)
    for (int idx = lane; idx < 16 * 3 * 64; idx += 32) {
        const int j  = idx & 63;
        const int rc = idx >> 6;      // 0..47
        const int c  = rc % 3;
        const int r  = rc / 3;
        int ci = x_cat[(size_t)(base + r) * 3 + c];
        ci = ci < 0 ? 0 : (ci > 10000 ? 10000 : ci);
        const float* tbl = (c == 0) ? emb0 : (c == 1) ? emb1 : emb2;
        feat[r][64 + c * 64 + j] = tbl[(size_t)ci * 64 + j];
    }
    __syncthreads();

    // ---- GEMM2: feat(16x256) @ w_in(256x128) -> h
    for (int nt = 0; nt < 8; ++nt) {
        v8f acc = {};
        const int ncol = nt * 16 + row16;
        for (int k = 0; k < FEAT; k += 4) {
            v2f a, b;
            a.x = feat[row16][k + khalf];
            a.y = feat[row16][k + khalf + 1];
            b.x = w_in[(k + khalf) * HIDDEN + ncol];
            b.y = w_in[(k + khalf + 1) * HIDDEN + ncol];
            acc = wmma_f32_k4(a, b, acc);
        }
        const float bias = b_in[ncol];
#pragma unroll
        for (int v = 0; v < 8; ++v) {
            float val = acc[v] + bias;
            h[(size_t)(base + crow0 + v) * HIDDEN + ncol] = val > 0.f ? val : 0.f;
        }
    }
}

// ---------------------------------------------------------------------------
// Kernel 2: zero agg (N*128) and deg (N)
// ---------------------------------------------------------------------------
__global__ __launch_bounds__(256) void gnn_zero_kernel(float* __restrict__ agg,
                                                       float* __restrict__ deg)
{
    const int i = blockIdx.x * blockDim.x + threadIdx.x;
    if (i < N_NODES * HIDDEN) agg[i] = 0.f;
    if (i < N_NODES)          deg[i] = 0.f;
}

// ---------------------------------------------------------------------------
// Kernel 3: edge scatter.  32 lanes per edge: each lane moves float4 (16 B),
// the wave covers one full 512 B feature row coalesced; f32 atomics to L2.
// ---------------------------------------------------------------------------
__global__ __launch_bounds__(256) void gnn_scatter_kernel(
    const int* __restrict__ src, const int* __restrict__ dst,
    const float* __restrict__ h, float* __restrict__ agg,
    float* __restrict__ deg)
{
    const int t = blockIdx.x * blockDim.x + threadIdx.x;
    const int e = t >> 5;
    const int lane = t & 31;
    if (e >= E_EDGES) return;
    const int s = src[e];
    const int d = dst[e];
    const float4 val = *(const float4*)(h + (size_t)s * HIDDEN + lane * 4);
    float* out = agg + (size_t)d * HIDDEN + lane * 4;
    atomicAdd(out + 0, val.x);
    atomicAdd(out + 1, val.y);
    atomicAdd(out + 2, val.z);
    atomicAdd(out + 3, val.w);
    if (lane == 0) atomicAdd(deg + d, 1.0f);
}

// ---------------------------------------------------------------------------
// Kernel 4: GCN update.  One wave per 16-node tile:
//   xs = agg/max(deg,1);  y = relu(xs @ W + b) + h;  h = layernorm(y)*g + bln
// ---------------------------------------------------------------------------
__global__ __launch_bounds__(32) void gnn_update_kernel(
    const float* __restrict__ agg, const float* __restrict__ deg,
    const float* __restrict__ w, const float* __restrict__ bvec,
    const float* __restrict__ g, const float* __restrict__ bln,
    float* __restrict__ h)
{
    __shared__ float xs[16][HIDDEN];
    __shared__ float xr[16][HIDDEN];
    __shared__ float stats[16][2];
    const int lane  = threadIdx.x;
    const int base  = blockIdx.x * 16;
    const int row16 = lane & 15;
    const int khalf = (lane >> 4) * 2;
    const int crow0 = (lane >> 4) * 8;

    // mean aggregation into LDS
    for (int idx = lane; idx < 16 * HIDDEN; idx += 32) {
        const int r = idx >> 7, c = idx & 127;
        float dg = deg[base + r];
        dg = dg < 1.f ? 1.f : dg;
        xs[r][c] = agg[(size_t)(base + r) * HIDDEN + c] / dg;
    }
    __syncthreads();

    // GEMM: xs(16x128) @ w(128x128), bias + relu + residual -> xr
    for (int nt = 0; nt < 8; ++nt) {
        v8f acc = {};
        const int ncol = nt * 16 + row16;
        for (int k = 0; k < HIDDEN; k += 4) {
            v2f a, b;
            a.x = xs[row16][k + khalf];
            a.y = xs[row16][k + khalf + 1];
            b.x = w[(k + khalf) * HIDDEN + ncol];
            b.y = w[(k + khalf + 1) * HIDDEN + ncol];
            acc = wmma_f32_k4(a, b, acc);
        }
        const float bias = bvec[ncol];
#pragma unroll
        for (int v = 0; v < 8; ++v) {
            const int r = crow0 + v;
            float val = acc[v] + bias;
            val = val > 0.f ? val : 0.f;
            xr[r][ncol] = val + h[(size_t)(base + r) * HIDDEN + ncol];
        }
    }
    __syncthreads();

    // per-row layernorm statistics
    if (lane < 16) {
        float mu = 0.f;
        for (int c = 0; c < HIDDEN; ++c) mu += xr[lane][c];
        mu *= (1.f / HIDDEN);
        float var = 0.f;
        for (int c = 0; c < HIDDEN; ++c) { const float d0 = xr[lane][c] - mu; var += d0 * d0; }
        var *= (1.f / HIDDEN);
        stats[lane][0] = mu;
        stats[lane][1] = rsqrtf(var + 1e-5f);
    }
    __syncthreads();

    for (int idx = lane; idx < 16 * HIDDEN; idx += 32) {
        const int r = idx >> 7, c = idx & 127;
        h[(size_t)(base + r) * HIDDEN + c] =
            (xr[r][c] - stats[r][0]) * stats[r][1] * g[c] + bln[c];
    }
}

// ---------------------------------------------------------------------------
// Kernel 5: output head.  One wave per node: coalesced float4 dot + shuffle
// reduction across 32 lanes.
// ---------------------------------------------------------------------------
__global__ __launch_bounds__(256) void gnn_out_kernel(
    const float* __restrict__ h, const float* __restrict__ w,
    const float* __restrict__ b, float* __restrict__ out)
{
    const int t = blockIdx.x * blockDim.x + threadIdx.x;
    const int n = t >> 5;
    const int lane = t & 31;
    if (n >= N_NODES) return;
    const float4 hv = *(const float4*)(h + (size_t)n * HIDDEN + lane * 4);
    const float4 wv = *(const float4*)(w + lane * 4);
    float s = hv.x * wv.x + hv.y * wv.y + hv.z * wv.z + hv.w * wv.w;
#pragma unroll
    for (int off = 16; off > 0; off >>= 1)
        s += __shfl_down(s, off, 32);
    if (lane == 0) out[n] = s + b[0];
}

extern "C" void kernel_launch(void* const* d_in, const int* in_sizes, int n_in,
                              void* d_out, int out_size, void* d_ws, size_t ws_size,
                              hipStream_t stream) {
    const float* x_num = (const float*)d_in[0];
    const int*   x_cat = (const int*)d_in[1];
    const int*   edge  = (const int*)d_in[2];
    const float* wnum  = (const float*)d_in[3];
    const float* bnum  = (const float*)d_in[4];
    const float* e0    = (const float*)d_in[5];
    const float* e1    = (const float*)d_in[6];
    const float* e2    = (const float*)d_in[7];
    const float* win   = (const float*)d_in[8];
    const float* bin   = (const float*)d_in[9];
    const float* w1    = (const float*)d_in[10];
    const float* b1    = (const float*)d_in[11];
    const float* w2    = (const float*)d_in[12];
    const float* b2    = (const float*)d_in[13];
    const float* g1    = (const float*)d_in[14];
    const float* bln1  = (const float*)d_in[15];
    const float* g2    = (const float*)d_in[16];
    const float* bln2  = (const float*)d_in[17];
    const float* wout  = (const float*)d_in[18];
    const float* bout  = (const float*)d_in[19];
    float* out = (float*)d_out;

    float* h   = (float*)d_ws;                      // N*128
    float* agg = h + (size_t)N_NODES * HIDDEN;      // N*128
    float* deg = agg + (size_t)N_NODES * HIDDEN;    // N

    const int* srcp = edge;
    const int* dstp = edge + E_EDGES;

    const int tiles   = N_NODES / 16;               // 3125 (exact)
    const int zgrid   = (N_NODES * HIDDEN + 255) / 256;
    const int sgrid   = (E_EDGES * 32) / 256;       // 100000 (exact)
    const int ogrid   = (N_NODES * 32 + 255) / 256;

    gnn_encode_kernel<<<tiles, 32, 0, stream>>>(x_num, x_cat, wnum, bnum,
                                                e0, e1, e2, win, bin, h);
    // block 1
    gnn_zero_kernel<<<zgrid, 256, 0, stream>>>(agg, deg);
    gnn_scatter_kernel<<<sgrid, 256, 0, stream>>>(srcp, dstp, h, agg, deg);
    gnn_update_kernel<<<tiles, 32, 0, stream>>>(agg, deg, w1, b1, g1, bln1, h);
    // block 2
    gnn_zero_kernel<<<zgrid, 256, 0, stream>>>(agg, deg);
    gnn_scatter_kernel<<<sgrid, 256, 0, stream>>>(srcp, dstp, h, agg, deg);
    gnn_update_kernel<<<tiles, 32, 0, stream>>>(agg, deg, w2, b2, g2, bln2, h);
    // head
    gnn_out_kernel<<<ogrid, 256, 0, stream>>>(h, wout, bout, out);
}